// HungarianMatcher_29489245454396
// MI455X (gfx1250) — compile-verified
//
#include <hip/hip_runtime.h>
#include <hip/hip_bf16.h>
#include <float.h>
#include <stdint.h>

#define W_CLASS 2.0f
#define W_BBOX  5.0f
#define W_GIOU  2.0f

static constexpr int TM = 256;   // targets per block == threads per block
static constexpr int NR = 64;    // query rows per block

// focal class cost diff: pos_cost - neg_cost for one logit
__device__ __forceinline__ float focal_diff(float x) {
    float ax = fabsf(x);
    float t  = log1pf(expf(-ax));          // log(1+e^-|x|)
    float sp_pos = fmaxf(x, 0.0f)  + t;    // softplus(x)  = -log_sigmoid(-x)
    float sp_neg = fmaxf(-x, 0.0f) + t;    // softplus(-x) = -log_sigmoid(x)
    float p   = 1.0f / (1.0f + expf(-x));  // sigmoid(x)
    float omp = 1.0f - p;
    float pos = 0.25f * omp * omp * sp_neg;
    float neg = 0.75f * p   * p   * sp_pos;
    return pos - neg;
}

__device__ __forceinline__ unsigned flip_f32(float f) {
    unsigned u = __float_as_uint(f);
    return (u & 0x80000000u) ? ~u : (u | 0x80000000u);
}
__device__ __forceinline__ float unflip_f32(unsigned su) {
    return (su & 0x80000000u) ? __uint_as_float(su & 0x7FFFFFFFu)
                              : __uint_as_float(~su);
}

__global__ void init_kernel(unsigned* maxscr) {
    if (threadIdx.x == 0) *maxscr = 0x007FFFFFu; // flip(-inf)
}

__global__ void focal_kernel(const float* __restrict__ logits,
                             float* __restrict__ tab, int total) {
    int i = blockIdx.x * blockDim.x + threadIdx.x;
    if (i < total) tab[i] = focal_diff(logits[i]);
}

// Precompute per-query xyxy box + area (reference formula: area from xyxy diffs)
__global__ void qprep_kernel(const float* __restrict__ pred_boxes,
                             float4* __restrict__ qxyxy,
                             float*  __restrict__ qarea, int N) {
    int i = blockIdx.x * blockDim.x + threadIdx.x;
    if (i < N) {
        float4 b = ((const float4*)pred_boxes)[i];
        float w = fmaxf(b.z, 0.0f), h = fmaxf(b.w, 0.0f);
        float x0 = b.x - 0.5f * w, y0 = b.y - 0.5f * h;
        float x1 = b.x + 0.5f * w, y1 = b.y + 0.5f * h;
        qxyxy[i] = make_float4(x0, y0, x1, y1);
        qarea[i] = (x1 - x0) * (y1 - y0);
    }
}

template <bool USE_TABLE>
__global__ __launch_bounds__(256)
void cost_kernel(const float*  __restrict__ cls_src,    // focal table OR raw logits (N,C)
                 const float*  __restrict__ pred_boxes, // (N,4) raw cxcywh
                 const float4* __restrict__ qxyxy,      // (N) xyxy   (table path)
                 const float*  __restrict__ qarea,      // (N) area   (table path)
                 const float*  __restrict__ tgt_bbox,   // (M,4) cxcywh
                 const int*    __restrict__ tgt_ids,    // (M,)
                 float* __restrict__ out,               // (N,M)
                 unsigned* __restrict__ maxscr,
                 int N, int M, int C)
{
    __shared__ float4 s_xy[NR];   // query xyxy
    __shared__ float4 s_rw[NR];   // query raw cxcywh
    __shared__ float  s_ar[NR];   // query area
    __shared__ float  redbuf[8];

    const int tid = threadIdx.x;
    const int m   = blockIdx.x * TM + tid;
    const int n0  = blockIdx.y * NR;
    const bool mvalid = (m < M);

    // ---- per-lane target box, amortized over NR rows ----
    float tbx = 0.f, tby = 0.f, tbz = 0.f, tbw = 0.f;
    float tx0 = 0.f, ty0 = 0.f, tx1 = 0.f, ty1 = 0.f, tarea = 0.f;
    int   id  = 0;
    if (mvalid) {
        float4 b = ((const float4*)tgt_bbox)[m];
        tbx = b.x; tby = b.y; tbz = b.z; tbw = b.w;
        float w = fmaxf(b.z, 0.0f), h = fmaxf(b.w, 0.0f);
        tx0 = b.x - 0.5f * w; ty0 = b.y - 0.5f * h;
        tx1 = b.x + 0.5f * w; ty1 = b.y + 0.5f * h;
        tarea = (tx1 - tx0) * (ty1 - ty0);
        id = tgt_ids[m];
    }

    // ---- gfx1250 async loads: stage this block's query data into LDS ----
    if (tid < NR && (n0 + tid) < N) {
        const int n = n0 + tid;
        {
            unsigned lo = (unsigned)(uintptr_t)(&s_rw[tid]);
            const float4* g = ((const float4*)pred_boxes) + n;
            asm volatile("global_load_async_to_lds_b128 %0, %1, off"
                         :: "v"(lo), "v"(g) : "memory");
        }
        if (USE_TABLE) {
            unsigned lo = (unsigned)(uintptr_t)(&s_xy[tid]);
            const float4* g = qxyxy + n;
            asm volatile("global_load_async_to_lds_b128 %0, %1, off"
                         :: "v"(lo), "v"(g) : "memory");
            unsigned lo2 = (unsigned)(uintptr_t)(&s_ar[tid]);
            const float* g2 = qarea + n;
            asm volatile("global_load_async_to_lds_b32 %0, %1, off"
                         :: "v"(lo2), "v"(g2) : "memory");
        }
    }
    asm volatile("s_wait_asynccnt 0" ::: "memory");
    __syncthreads();

    float lmax = -INFINITY;
    const int nr = min(NR, N - n0);
    float* orow = out + (size_t)n0 * M + m;
    size_t cbase = (size_t)n0 * C + id;

    #pragma unroll 2
    for (int r = 0; r < nr; ++r) {
        float4 q  = s_rw[r];                      // LDS broadcast b128

        float qx0, qy0, qx1, qy1, qa;
        if (USE_TABLE) {
            float4 xy = s_xy[r];
            qx0 = xy.x; qy0 = xy.y; qx1 = xy.z; qy1 = xy.w;
            qa  = s_ar[r];
        } else {
            float w = fmaxf(q.z, 0.0f), h = fmaxf(q.w, 0.0f);
            qx0 = q.x - 0.5f * w; qy0 = q.y - 0.5f * h;
            qx1 = q.x + 0.5f * w; qy1 = q.y + 0.5f * h;
            qa  = (qx1 - qx0) * (qy1 - qy0);
        }

        // intersection / union / iou (fast rcp: non-finite propagates for fixup)
        float iw = fmaxf(fminf(qx1, tx1) - fmaxf(qx0, tx0), 0.0f);
        float ih = fmaxf(fminf(qy1, ty1) - fmaxf(qy0, ty0), 0.0f);
        float inter = iw * ih;
        float uni   = qa + tarea - inter;
        float iou   = inter * __builtin_amdgcn_rcpf(uni);

        // enclosing box / giou
        float ew = fmaxf(fmaxf(qx1, tx1) - fminf(qx0, tx0), 0.0f);
        float eh = fmaxf(fmaxf(qy1, ty1) - fminf(qy0, ty0), 0.0f);
        float earea = ew * eh;
        float giou  = iou - (earea - uni) * __builtin_amdgcn_rcpf(earea);

        // L1 on raw cxcywh
        float l1 = fabsf(q.x - tbx) + fabsf(q.y - tby)
                 + fabsf(q.z - tbz) + fabsf(q.w - tbw);

        // class cost gather
        float cls;
        if (USE_TABLE) cls = cls_src[cbase];
        else           cls = focal_diff(cls_src[cbase]);
        cbase += C;

        float cost = fmaf(W_BBOX, l1, fmaf(W_CLASS, cls, -W_GIOU * giou));

        if (mvalid) {
            *orow = cost;
            if (fabsf(cost) <= FLT_MAX)            // finite (NaN fails compare)
                lmax = fmaxf(lmax, cost);
        }
        orow += M;
    }

    // ---- wave32 + block reduction of the finite max ----
    for (int o = 16; o > 0; o >>= 1)
        lmax = fmaxf(lmax, __shfl_xor(lmax, o, 32));
    if ((tid & 31) == 0) redbuf[tid >> 5] = lmax;
    __syncthreads();
    if (tid == 0) {
        float bm = redbuf[0];
        #pragma unroll
        for (int i = 1; i < 8; ++i) bm = fmaxf(bm, redbuf[i]);
        atomicMax(maxscr, flip_f32(bm));
    }
}

__global__ __launch_bounds__(256)
void fixup_kernel(float* __restrict__ out, const unsigned* __restrict__ maxscr,
                  size_t total)
{
    size_t i = ((size_t)blockIdx.x * blockDim.x + threadIdx.x) * 4;
    if (i >= total) return;
    float rep = 2.0f * unflip_f32(*maxscr);
    if (i + 4 <= total) {
        float4 v = *(float4*)(out + i);
        bool b0 = !(fabsf(v.x) <= FLT_MAX);
        bool b1 = !(fabsf(v.y) <= FLT_MAX);
        bool b2 = !(fabsf(v.z) <= FLT_MAX);
        bool b3 = !(fabsf(v.w) <= FLT_MAX);
        if (b0 | b1 | b2 | b3) {
            if (b0) v.x = rep;
            if (b1) v.y = rep;
            if (b2) v.z = rep;
            if (b3) v.w = rep;
            *(float4*)(out + i) = v;
        }
    } else {
        for (size_t k = i; k < total; ++k) {
            float v = out[k];
            if (!(fabsf(v) <= FLT_MAX)) out[k] = rep;
        }
    }
}

static inline size_t align256(size_t x) { return (x + 255) & ~(size_t)255; }

extern "C" void kernel_launch(void* const* d_in, const int* in_sizes, int n_in,
                              void* d_out, int out_size, void* d_ws, size_t ws_size,
                              hipStream_t stream) {
    const float* pred_logits = (const float*)d_in[0];
    const float* pred_boxes  = (const float*)d_in[1];
    const float* tgt_bbox    = (const float*)d_in[2];
    const int*   tgt_ids     = (const int*)d_in[3];
    float* out = (float*)d_out;

    const int M = in_sizes[3];          // 2048
    const int N = in_sizes[1] / 4;      // 12000
    const int C = in_sizes[0] / N;      // 91

    // workspace layout
    unsigned* maxscr = (unsigned*)d_ws;
    size_t off = 256;
    float*  tab   = (float*)((char*)d_ws + off);  off += align256((size_t)N * C * 4);
    float4* qxyxy = (float4*)((char*)d_ws + off); off += align256((size_t)N * 16);
    float*  qarea = (float*)((char*)d_ws + off);  off += align256((size_t)N * 4);
    const bool useTable = (ws_size >= off);

    init_kernel<<<1, 32, 0, stream>>>(maxscr);

    dim3 grid((M + TM - 1) / TM, (N + NR - 1) / NR);
    if (useTable) {
        const int tot = N * C;
        focal_kernel<<<(tot + 255) / 256, 256, 0, stream>>>(pred_logits, tab, tot);
        qprep_kernel<<<(N + 255) / 256, 256, 0, stream>>>(pred_boxes, qxyxy, qarea, N);
        cost_kernel<true><<<grid, 256, 0, stream>>>(tab, pred_boxes, qxyxy, qarea,
                                                    tgt_bbox, tgt_ids, out, maxscr,
                                                    N, M, C);
    } else {
        cost_kernel<false><<<grid, 256, 0, stream>>>(pred_logits, pred_boxes,
                                                     (const float4*)nullptr,
                                                     (const float*)nullptr,
                                                     tgt_bbox, tgt_ids, out, maxscr,
                                                     N, M, C);
    }

    const size_t total = (size_t)N * M;
    const size_t nblk = (total / 4 + 255) / 256;
    fixup_kernel<<<nblk, 256, 0, stream>>>(out, maxscr, total);
}